// PostProcessor_87213605913102
// MI455X (gfx1250) — compile-verified
//
#include <hip/hip_runtime.h>
#include <hip/hip_bf16.h>

// ---------------- problem constants ----------------
#define Bn   32
#define Cc   3
#define Hh   192
#define Ww   640
#define HWsz (Hh*Ww)
#define TOPKN 50
#define NDET (Bn*TOPKN)          // 1600
#define NSLICE (Bn*Cc)           // 96

// radix-select histogram over float bits (values in [0,1))
#define NBUCK   2049             // bucket 0 = (0,0.5); 1..2048 = [0.5,1.0) fine bins
#define HISTPAD 2052
#define CANDMAX 512

// NMS tiling: 16 output rows per block, 18 input rows in LDS
#define OROWS 16
#define CHUNKS (Hh/OROWS)        // 12
#define TROWS (OROWS+2)          // 18

// workspace layout (in 4-byte elements)
#define OFF_HIST 0
#define OFF_CCNT (OFF_HIST + NSLICE*HISTPAD)          // 196992
#define OFF_T    (OFF_CCNT + NSLICE)                  // 197088
#define OFF_CIND (OFF_T    + NSLICE)                  // 197184
#define OFF_CSC  (OFF_CIND + NSLICE*CANDMAX)          // 246336
#define OFF_TIND (OFF_CSC  + NSLICE*CANDMAX)          // 295488
#define OFF_TSC  (OFF_TIND + NSLICE*TOPKN)            // 300288
#define OFF_DET  (OFF_TSC  + NSLICE*TOPKN)            // 305088  (NDET*12 floats)

#define PI_F 3.14159265358979f
#define DET_THRESH 0.25f

typedef __attribute__((ext_vector_type(2))) float v2f;
typedef __attribute__((ext_vector_type(8))) float v8f;

__device__ __constant__ float cSX[8] = {-0.5f, 0.5f, 0.5f, 0.5f, 0.5f,-0.5f,-0.5f,-0.5f};
__device__ __constant__ float cSY[8] = {-1.0f,-1.0f, 0.0f, 0.0f,-1.0f,-1.0f, 0.0f, 0.0f};
__device__ __constant__ float cSZ[8] = {-0.5f,-0.5f,-0.5f, 0.5f, 0.5f, 0.5f, 0.5f,-0.5f};
__device__ __constant__ float cDIM[3][3] = {{3.88f,1.63f,1.53f},{0.84f,1.76f,0.66f},{1.78f,1.52f,1.78f}};

__device__ __forceinline__ int bucket_of(float v) {
  if (v >= 0.5f) {
    unsigned bits = __float_as_uint(v);
    int b = 1 + (int)((bits - 0x3F000000u) >> 12);
    return (b > 2048) ? 2048 : b;
  }
  return 0;
}

__device__ __forceinline__ float wrapang(float a) {
  if (a >  PI_F) a -= 2.0f * PI_F;
  else if (a < -PI_F) a += 2.0f * PI_F;
  return a;
}

// ---- CDNA5 async global->LDS tile staging (ASYNCcnt path) ----
// Copies rows [r0-1, r0+OROWS] of slice `slice` of the heatmap into `tile`
// (TROWS x Ww floats); out-of-range rows pre-filled with -1e30f.
__device__ __forceinline__ void load_tile_async(const float* __restrict__ heat,
                                                float* tile, int slice, int r0, int tid) {
  for (int tr = 0; tr < TROWS; ++tr) {
    int gr = r0 - 1 + tr;
    if (gr < 0 || gr >= Hh) {
      for (int x = tid; x < Ww; x += 256) tile[tr * Ww + x] = -1e30f;
    }
  }
  unsigned ldsbase  = (unsigned)(size_t)tile;              // low 32 bits = LDS byte addr
  unsigned sliceoff = (unsigned)slice * (unsigned)(HWsz * 4);
  for (int t = tid; t < TROWS * (Ww / 4); t += 256) {
    int tr = t / (Ww / 4);
    int c4 = (t % (Ww / 4)) * 4;
    int gr = r0 - 1 + tr;
    if (gr >= 0 && gr < Hh) {
      unsigned goff = sliceoff + (unsigned)(gr * Ww + c4) * 4u;
      unsigned loff = ldsbase + (unsigned)(tr * Ww + c4) * 4u;
      asm volatile("global_load_async_to_lds_b128 %0, %1, %2"
                   :: "v"(loff), "v"(goff), "s"(heat) : "memory");
    }
  }
  asm volatile("s_wait_asynccnt 0" ::: "memory");
  __syncthreads();
}

__device__ __forceinline__ float nms_at(const float* tile, int tr, int x) {
  float h = tile[tr * Ww + x];
  float mv = h;
  #pragma unroll
  for (int dr = -1; dr <= 1; ++dr) {
    #pragma unroll
    for (int dc = -1; dc <= 1; ++dc) {
      int xx = x + dc;
      if (xx >= 0 && xx < Ww) mv = fmaxf(mv, tile[(tr + dr) * Ww + xx]);
    }
  }
  return (mv <= h) ? h : 0.0f;   // mv>=h always; equal iff h is the 3x3 max
}

// ---------------- kernel 1: NMS + per-slice histogram ----------------
__global__ void k_nms_hist(const float* __restrict__ heat, int* __restrict__ ghist) {
  __shared__ float tile[TROWS * Ww];
  __shared__ int lhist[HISTPAD];
  int slice = blockIdx.x;
  int r0 = blockIdx.y * OROWS;
  int tid = threadIdx.x;
  for (int i = tid; i < HISTPAD; i += 256) lhist[i] = 0;
  load_tile_async(heat, tile, slice, r0, tid);
  for (int t = tid; t < OROWS * Ww; t += 256) {
    int orow = t / Ww, x = t % Ww;
    float nms = nms_at(tile, orow + 1, x);
    if (nms > 0.0f) atomicAdd(&lhist[bucket_of(nms)], 1);
  }
  __syncthreads();
  for (int i = tid; i < NBUCK; i += 256) {
    int v = lhist[i];
    if (v) atomicAdd(&ghist[slice * HISTPAD + i], v);
  }
}

// ---------------- kernel 2: per-slice threshold bucket ----------------
__global__ void k_thresh(const int* __restrict__ ghist, int* __restrict__ sliceT) {
  int slice = blockIdx.x;
  if (threadIdx.x == 0) {
    int cum = 0, T = 0;
    for (int bkt = NBUCK - 1; bkt >= 0; --bkt) {
      cum += ghist[slice * HISTPAD + bkt];
      if (cum >= TOPKN) { T = bkt; break; }
    }
    sliceT[slice] = T;
  }
}

// ---------------- kernel 3: second pass, collect candidates ----------------
__global__ void k_nms_collect(const float* __restrict__ heat, const int* __restrict__ sliceT,
                              int* __restrict__ candCnt, int* __restrict__ candInd,
                              float* __restrict__ candSc) {
  __shared__ float tile[TROWS * Ww];
  int slice = blockIdx.x;
  int r0 = blockIdx.y * OROWS;
  int tid = threadIdx.x;
  int T = sliceT[slice];
  load_tile_async(heat, tile, slice, r0, tid);
  for (int t = tid; t < OROWS * Ww; t += 256) {
    int orow = t / Ww, x = t % Ww;
    float nms = nms_at(tile, orow + 1, x);
    if (nms > 0.0f && bucket_of(nms) >= T) {
      int p = atomicAdd(&candCnt[slice], 1);
      if (p < CANDMAX) {
        candSc[slice * CANDMAX + p] = nms;
        candInd[slice * CANDMAX + p] = (r0 + orow) * Ww + x;
      }
    }
  }
}

// ---------------- kernel 4: exact per-slice top-50 (rank by count) ----------------
__global__ void k_slice_topk(const int* __restrict__ candCnt, const int* __restrict__ candInd,
                             const float* __restrict__ candSc,
                             float* __restrict__ topSc, int* __restrict__ topInd) {
  __shared__ float ss[CANDMAX];
  __shared__ int si[CANDMAX];
  int slice = blockIdx.x;
  int tid = threadIdx.x;      // 512 threads
  int n = candCnt[slice]; if (n > CANDMAX) n = CANDMAX;
  if (tid < n) { ss[tid] = candSc[slice * CANDMAX + tid]; si[tid] = candInd[slice * CANDMAX + tid]; }
  if (tid < TOPKN) { topSc[slice * TOPKN + tid] = 0.0f; topInd[slice * TOPKN + tid] = 0; }
  __syncthreads();
  if (tid < n) {
    float s = ss[tid]; int id = si[tid]; int rank = 0;
    for (int j = 0; j < n; ++j) {
      float sj = ss[j];
      rank += (sj > s) || (sj == s && si[j] < id);   // tie-break = flat index (lax.top_k order)
    }
    if (rank < TOPKN) { topSc[slice * TOPKN + rank] = s; topInd[slice * TOPKN + rank] = id; }
  }
}

__device__ __forceinline__ void inv3(const float m[9], float o[9]) {
  float a=m[0],b=m[1],c=m[2],d=m[3],e=m[4],f=m[5],g=m[6],h=m[7],i=m[8];
  float A = e*i - f*h, Bc = -(d*i - f*g), Cc2 = d*h - e*g;
  float det = a*A + b*Bc + c*Cc2;
  float inv = 1.0f / det;
  o[0]=A*inv;             o[1]=-(b*i-c*h)*inv;  o[2]=(b*f-c*e)*inv;
  o[3]=Bc*inv;            o[4]=(a*i-c*g)*inv;   o[5]=-(a*f-c*d)*inv;
  o[6]=Cc2*inv;           o[7]=-(a*h-b*g)*inv;  o[8]=(a*e-b*d)*inv;
}

// ---------------- kernel 5: per-batch merge + geometry params ----------------
__global__ void k_batch(const float* __restrict__ topSc, const int* __restrict__ topInd,
                        const float* __restrict__ preg, const float* __restrict__ trans,
                        const float* __restrict__ Kmat, float* __restrict__ det) {
  __shared__ float s150[160];
  __shared__ int   i150[160];
  __shared__ float selS[TOPKN];
  __shared__ int   selI[TOPKN];
  __shared__ int   selC[TOPKN];
  int b = blockIdx.x;
  int tid = threadIdx.x;     // 256 threads
  if (tid < Cc * TOPKN) {
    int c = tid / TOPKN, k = tid % TOPKN;
    int slice = b * Cc + c;
    s150[tid] = topSc[slice * TOPKN + k];
    i150[tid] = topInd[slice * TOPKN + k];
  }
  __syncthreads();
  if (tid < Cc * TOPKN) {
    float s = s150[tid]; int rank = 0;
    for (int j = 0; j < Cc * TOPKN; ++j) {
      float sj = s150[j];
      rank += (sj > s) || (sj == s && j < tid);      // concatenation order = c*K+k
    }
    if (rank < TOPKN) { selS[rank] = s; selI[rank] = i150[tid]; selC[rank] = tid / TOPKN; }
  }
  __syncthreads();
  if (tid < TOPKN) {
    float score = selS[tid];
    int ind = selI[tid];
    int c = selC[tid];
    float xs = (float)(ind % Ww), ys = (float)(ind / Ww);
    const float* pr = preg + (size_t)b * 8 * HWsz;
    float r0 = pr[0*HWsz+ind], r1 = pr[1*HWsz+ind], r2 = pr[2*HWsz+ind], r3 = pr[3*HWsz+ind];
    float r4 = pr[4*HWsz+ind], r5 = pr[5*HWsz+ind], r6 = pr[6*HWsz+ind], r7 = pr[7*HWsz+ind];
    float tm[9], km[9], ti[9], ki[9];
    #pragma unroll
    for (int t = 0; t < 9; ++t) { tm[t] = trans[b*9+t]; km[t] = Kmat[b*9+t]; }
    inv3(tm, ti); inv3(km, ki);
    float px = xs + r1, py = ys + r2;
    float p0 = ti[0]*px + ti[1]*py + ti[2];
    float p1 = ti[3]*px + ti[4]*py + ti[5];
    float p2 = ti[6]*px + ti[7]*py + ti[8];
    float depth = r0 * 16.32f + 28.01f;
    p0 *= depth; p1 *= depth; p2 *= depth;
    float lx = ki[0]*p0 + ki[1]*p1 + ki[2]*p2;
    float ly = ki[3]*p0 + ki[4]*p1 + ki[5]*p2;
    float lz = ki[6]*p0 + ki[7]*p1 + ki[8]*p2;
    float dm0 = expf(r3) * cDIM[c][0];
    float dm1 = expf(r4) * cDIM[c][1];
    float dm2 = expf(r5) * cDIM[c][2];
    ly += dm1 * 0.5f;
    float rays  = atanf(lx / (lz + 1e-7f));
    float alpha = atanf(r6 / (r7 + 1e-7f)) + ((r7 >= 0.0f) ? -0.5f*PI_F : 0.5f*PI_F);
    float roty  = alpha + rays;
    roty = wrapang(roty); alpha = wrapang(alpha);
    float* dp = det + (size_t)(b * TOPKN + tid) * 12;
    dp[0] = score; dp[1] = (float)c; dp[2] = alpha; dp[3] = roty;
    dp[4] = dm0; dp[5] = dm1; dp[6] = dm2;
    dp[7] = lx;  dp[8] = ly;  dp[9] = lz;
    dp[10] = 0.0f; dp[11] = 0.0f;
  }
}

// ---------------- kernel 6: WMMA corner projection + final output ----------------
// One wave per detection. Two V_WMMA_F32_16X16X4_F32:
//   D1 = RY(3x3 in 16x4) x corners(3x8 in 4x16);  D2 = K(3x3) x (D1 + locs)
// A layout (ISA 7.12.2, 32-bit A 16x4): lane m (0..15), VGPR0={K0|K2}, VGPR1={K1|K3}.
// B layout (row-striped): VGPR0 = row K0 (lanes 0-15) / row K2 (lanes 16-31),
//                         VGPR1 = row K1 / row K3.
// C/D: VGPR r, lane n -> element (M = r + 8*(lane>=16), N = n&15).
__global__ void k_box(const float* __restrict__ det, const float* __restrict__ Kmat,
                      const float* __restrict__ imgsz, float* __restrict__ out) {
  int gt = blockIdx.x * blockDim.x + threadIdx.x;
  int wid = gt >> 5;            // detection 0..1599 (grid sized exactly)
  int lane = gt & 31;
  const float* dp = det + (size_t)wid * 12;
  float score = dp[0], clsf = dp[1], alpha = dp[2], roty = dp[3];
  float dm0 = dp[4], dm1 = dp[5], dm2 = dp[6];
  float lx = dp[7], ly = dp[8], lz = dp[9];
  int b = wid / TOPKN;
  const float* km = Kmat + (size_t)b * 9;

  float co = cosf(roty), si = sinf(roty);
  int m = lane & 15;
  bool hi = lane >= 16;

  // corners (column m of B1), zero for m>=8
  float cx = (m < 8) ? dm0 * cSX[m & 7] : 0.0f;
  float cy = (m < 8) ? dm1 * cSY[m & 7] : 0.0f;
  float cz = (m < 8) ? dm2 * cSZ[m & 7] : 0.0f;

  // A1 = RY rows [c 0 s; 0 1 0; -s 0 c], padded
  float ax = (!hi) ? ((m == 0) ? co : ((m == 2) ? -si : 0.0f))
                   : ((m == 0) ? si : ((m == 2) ?  co : 0.0f));
  float ay = (!hi) ? ((m == 1) ? 1.0f : 0.0f) : 0.0f;
  v2f a1; a1.x = ax; a1.y = ay;
  v2f b1; b1.x = hi ? cz : cx; b1.y = hi ? 0.0f : cy;
  v8f z8 = {0.f,0.f,0.f,0.f,0.f,0.f,0.f,0.f};
  v8f d1 = __builtin_amdgcn_wmma_f32_16x16x4_f32(false, a1, false, b1, (short)0, z8, false, false);

  // camera-space corners (lanes 0..7 hold their corner's x,y,z in d1[0..2])
  bool cv = (m < 8);
  float xc = cv ? (d1[0] + lx) : 0.0f;
  float yc = cv ? (d1[1] + ly) : 0.0f;
  float zc = cv ? (d1[2] + lz) : 0.0f;
  float xcs = __shfl(xc, m, 32);
  float ycs = __shfl(yc, m, 32);
  float zcs = __shfl(zc, m, 32);

  // A2 = K (3x3) padded
  float a2x = (!hi) ? ((m < 3) ? km[m*3 + 0] : 0.0f) : ((m < 3) ? km[m*3 + 2] : 0.0f);
  float a2y = (!hi) ? ((m < 3) ? km[m*3 + 1] : 0.0f) : 0.0f;
  v2f a2; a2.x = a2x; a2.y = a2y;
  v2f b2; b2.x = hi ? zcs : xcs; b2.y = hi ? 0.0f : ycs;
  v8f d2 = __builtin_amdgcn_wmma_f32_16x16x4_f32(false, a2, false, b2, (short)0, z8, false, false);

  float w = d2[2];
  float u = d2[0] / w, v = d2[1] / w;   // valid for lanes 0..7

  float xmin = 1e30f, xmax = -1e30f, ymin = 1e30f, ymax = -1e30f;
  #pragma unroll
  for (int j = 0; j < 8; ++j) {
    float uj = __shfl(u, j, 32), vj = __shfl(v, j, 32);
    xmin = fminf(xmin, uj); xmax = fmaxf(xmax, uj);
    ymin = fminf(ymin, vj); ymax = fmaxf(ymax, vj);
  }
  if (lane == 0) {
    float Wi = imgsz[b*2 + 0], Hi = imgsz[b*2 + 1];
    xmin = fminf(fmaxf(xmin, 0.0f), Wi); xmax = fminf(fmaxf(xmax, 0.0f), Wi);
    ymin = fminf(fmaxf(ymin, 0.0f), Hi); ymax = fminf(fmaxf(ymax, 0.0f), Hi);
    bool keep = score > DET_THRESH;
    float* o = out + (size_t)wid * 14;
    o[0]  = keep ? clsf  : 0.0f;
    o[1]  = keep ? alpha : 0.0f;
    o[2]  = keep ? xmin  : 0.0f;
    o[3]  = keep ? ymin  : 0.0f;
    o[4]  = keep ? xmax  : 0.0f;
    o[5]  = keep ? ymax  : 0.0f;
    o[6]  = keep ? dm1   : 0.0f;   // dims rolled -1: [d1, d2, d0]
    o[7]  = keep ? dm2   : 0.0f;
    o[8]  = keep ? dm0   : 0.0f;
    o[9]  = keep ? lx    : 0.0f;
    o[10] = keep ? ly    : 0.0f;
    o[11] = keep ? lz    : 0.0f;
    o[12] = keep ? roty  : 0.0f;
    o[13] = keep ? score : 0.0f;
  }
}

// ---------------- launcher ----------------
extern "C" void kernel_launch(void* const* d_in, const int* in_sizes, int n_in,
                              void* d_out, int out_size, void* d_ws, size_t ws_size,
                              hipStream_t stream) {
  const float* heat  = (const float*)d_in[0];   // (32,3,192,640)
  const float* preg  = (const float*)d_in[1];   // (32,8,192,640)
  const float* trans = (const float*)d_in[2];   // (32,3,3)
  const float* Kmat  = (const float*)d_in[3];   // (32,3,3)
  const float* imgsz = (const float*)d_in[4];   // (32,2)
  float* out = (float*)d_out;                   // (1600,14)

  int*   wsi = (int*)d_ws;
  float* wsf = (float*)d_ws;
  int*   hist = wsi + OFF_HIST;
  int*   ccnt = wsi + OFF_CCNT;
  int*   sT   = wsi + OFF_T;
  int*   cind = wsi + OFF_CIND;
  float* csc  = wsf + OFF_CSC;
  int*   tind = wsi + OFF_TIND;
  float* tsc  = wsf + OFF_TSC;
  float* det  = wsf + OFF_DET;

  // zero histogram + candidate counters (contiguous region)
  hipMemsetAsync(hist, 0, (size_t)(NSLICE * HISTPAD + NSLICE) * sizeof(int), stream);

  dim3 gTile(NSLICE, CHUNKS);
  k_nms_hist   <<<gTile, 256, 0, stream>>>(heat, hist);
  k_thresh     <<<NSLICE, 32, 0, stream>>>(hist, sT);
  k_nms_collect<<<gTile, 256, 0, stream>>>(heat, sT, ccnt, cind, csc);
  k_slice_topk <<<NSLICE, 512, 0, stream>>>(ccnt, cind, csc, tsc, tind);
  k_batch      <<<Bn, 256, 0, stream>>>(tsc, tind, preg, trans, Kmat, det);
  k_box        <<<(NDET * 32) / 256, 256, 0, stream>>>(det, Kmat, imgsz, out);
}